// AxialAttention_79817672229051
// MI455X (gfx1250) — compile-verified
//
#include <hip/hip_runtime.h>

typedef __attribute__((ext_vector_type(2))) float v2f;
typedef __attribute__((ext_vector_type(8))) float v8f;

#define CDIM   256
#define HDIM   40
#define SP     1600        // W*D sites per h-slice
#define STOT   64000       // H*W*D
#define DP     41          // padded D stride in LDS (odd -> bank-conflict free)
#define CH_STRIDE (40 * DP)            // floats per channel tile (1640)
#define MAT_STRIDE (16 * CH_STRIDE)    // floats per q/k/v LDS array (26240)

__device__ __forceinline__ v8f wmma4(v2f a, v2f b, v8f c) {
    // D = A(16x4 f32) * B(4x16 f32) + C(16x16 f32)
    return __builtin_amdgcn_wmma_f32_16x16x4_f32(false, a, false, b, (short)0, c, false, false);
}

// ---------------------------------------------------------------------------
// Kernel 1: fused QKV projection (WMMA) + axial attention per channel.
// One block handles (b, h, cblk): 16 output channels x 1600 sites.
// q/k/v slabs live entirely in LDS (3 x 16 x 40 x 41 floats = 314880 B).
// Writes 3*attn_out (the three identical branches summed) to d_ws.
// ---------------------------------------------------------------------------
__global__ void __launch_bounds__(256) qkv_attn_kernel(
    const float* __restrict__ x,
    const float* __restrict__ wq, const float* __restrict__ bq,
    const float* __restrict__ wk, const float* __restrict__ bk,
    const float* __restrict__ wv, const float* __restrict__ bv,
    float* __restrict__ attn3)
{
    extern __shared__ float smem[];
    float* qs = smem;
    float* ks = smem + MAT_STRIDE;
    float* vs = smem + 2 * MAT_STRIDE;

    const int bid  = blockIdx.x;
    const int b    = bid / (HDIM * 16);
    const int rem  = bid % (HDIM * 16);
    const int h    = rem / 16;
    const int cblk = rem % 16;
    const int o0   = cblk * 16;

    const int lane  = threadIdx.x & 31;
    const int wid   = threadIdx.x >> 5;
    const int mhalf = lane >> 4;     // 0|1
    const int nlo   = lane & 15;
    const int kb    = 2 * mhalf;     // K sub-offset for A/B layout

    const float* xb = x + (size_t)b * CDIM * STOT + h * SP;

    // ---- GEMM phase: M=16 (out ch), N=1600 (sites), K=256; q,k,v share B ----
    for (int nt = wid; nt < 100; nt += 8) {
        const int site = nt * 16 + nlo;
        v8f accq = {}; v8f acck = {}; v8f accv = {};
        const float* xcol = xb + site;
        const float* wqr = wq + (o0 + nlo) * CDIM;
        const float* wkr = wk + (o0 + nlo) * CDIM;
        const float* wvr = wv + (o0 + nlo) * CDIM;
        #pragma unroll 4
        for (int k = 0; k < CDIM; k += 4) {
            const int c0 = k + kb;
            v2f a_q = *(const v2f*)(wqr + c0);
            v2f a_k = *(const v2f*)(wkr + c0);
            v2f a_v = *(const v2f*)(wvr + c0);
            v2f bb;
            bb.x = xcol[(size_t)c0 * STOT];
            bb.y = xcol[(size_t)(c0 + 1) * STOT];
            accq = wmma4(a_q, bb, accq);
            acck = wmma4(a_k, bb, acck);
            accv = wmma4(a_v, bb, accv);
        }
        const int wc = site / 40;
        const int dc = site % 40;
        #pragma unroll
        for (int r = 0; r < 8; ++r) {
            const int m  = r + 8 * mhalf;
            const int oc = o0 + m;
            const int li = (m * 40 + wc) * DP + dc;
            qs[li] = accq[r] + bq[oc];
            ks[li] = acck[r] + bk[oc];
            vs[li] = accv[r] + bv[oc];
        }
    }
    __syncthreads();

    // ---- Attention phase: each wave handles 2 channels; S=40 (w), F=40 (d) ----
    const float scale = 0.17677669529663687f;   // 1/sqrt(hd=32)
    for (int cc = 0; cc < 2; ++cc) {
        const int ch = wid * 2 + cc;
        const float* qc = qs + ch * CH_STRIDE;
        const float* kc = ks + ch * CH_STRIDE;
        const float* vc = vs + ch * CH_STRIDE;
        float* obase = attn3 + (size_t)(b * CDIM + o0 + ch) * STOT + h * SP;

        const float* k0 = kc + lane * DP;              // j = lane      (0..31)
        const float* k1 = kc + (32 + (lane & 7)) * DP; // j = 32+lane   (lanes 0..7)

        for (int i = 0; i < 40; ++i) {
            const float* qrow = qc + i * DP;
            float s0 = 0.f, s1 = 0.f;
            #pragma unroll
            for (int d = 0; d < 40; ++d) {
                const float qv = qrow[d];
                s0 = __builtin_fmaf(qv, k0[d], s0);
                s1 = __builtin_fmaf(qv, k1[d], s1);
            }
            s0 *= scale; s1 *= scale;

            float smax = (lane < 8) ? fmaxf(s0, s1) : s0;
            #pragma unroll
            for (int off = 16; off > 0; off >>= 1)
                smax = fmaxf(smax, __shfl_xor(smax, off, 32));

            const float e0 = __expf(s0 - smax);
            const float e1 = (lane < 8) ? __expf(s1 - smax) : 0.f;
            float tsum = e0 + e1;
            #pragma unroll
            for (int off = 16; off > 0; off >>= 1)
                tsum += __shfl_xor(tsum, off, 32);
            const float inv = 1.0f / tsum;
            const float p0 = e0 * inv;
            const float p1 = e1 * inv;

            float a0 = 0.f, a1 = 0.f;
            #pragma unroll
            for (int j = 0; j < 40; ++j) {
                const float pj = (j < 32) ? __shfl(p0, j, 32) : __shfl(p1, j - 32, 32);
                const float* vrow = vc + j * DP;
                a0 = __builtin_fmaf(pj, vrow[lane], a0);
                a1 = __builtin_fmaf(pj, vrow[32 + (lane & 7)], a1);
            }
            float* orow = obase + i * 40;
            orow[lane] = 3.0f * a0;                 // three identical branches
            if (lane < 8) orow[32 + lane] = 3.0f * a1;
        }
    }
}

// ---------------------------------------------------------------------------
// Kernel 2: final 1x1x1 projection. y = wp @ attn3 + bp.
// One wave = one 16x16 output tile; K=256 via v_wmma_f32_16x16x4_f32.
// ---------------------------------------------------------------------------
__global__ void __launch_bounds__(256) proj_kernel(
    const float* __restrict__ a3,
    const float* __restrict__ wp, const float* __restrict__ bp,
    float* __restrict__ out)
{
    const int lane  = threadIdx.x & 31;
    const int wid   = threadIdx.x >> 5;
    const int t     = blockIdx.x * 8 + wid;     // 128000 tiles total
    const int nt    = t % 4000;
    const int mt    = (t / 4000) % 16;
    const int b     = t / 64000;
    const int o0    = mt * 16;
    const int mhalf = lane >> 4;
    const int nlo   = lane & 15;
    const int kb    = 2 * mhalf;
    const int s0    = nt * 16 + nlo;

    const float* ab  = a3 + (size_t)b * CDIM * STOT + s0;
    const float* wpr = wp + (o0 + nlo) * CDIM;

    v8f acc = {};
    #pragma unroll 4
    for (int k = 0; k < CDIM; k += 4) {
        const int c0 = k + kb;
        v2f aa = *(const v2f*)(wpr + c0);
        v2f bb;
        bb.x = ab[(size_t)c0 * STOT];
        bb.y = ab[(size_t)(c0 + 1) * STOT];
        acc = wmma4(aa, bb, acc);
    }
    #pragma unroll
    for (int r = 0; r < 8; ++r) {
        const int m = r + 8 * mhalf;
        out[(size_t)(b * CDIM + o0 + m) * STOT + s0] = acc[r] + bp[o0 + m];
    }
}

extern "C" void kernel_launch(void* const* d_in, const int* in_sizes, int n_in,
                              void* d_out, int out_size, void* d_ws, size_t ws_size,
                              hipStream_t stream) {
    const float* x  = (const float*)d_in[0];
    const float* wq = (const float*)d_in[1];
    const float* bq = (const float*)d_in[2];
    const float* wk = (const float*)d_in[3];
    const float* bk = (const float*)d_in[4];
    const float* wv = (const float*)d_in[5];
    const float* bv = (const float*)d_in[6];
    const float* wp = (const float*)d_in[7];
    const float* bp = (const float*)d_in[8];

    float* attn3 = (float*)d_ws;   // needs 2*256*64000*4 = 131,072,000 bytes
    float* out   = (float*)d_out;

    const size_t smem = (size_t)(3 * MAT_STRIDE) * sizeof(float);  // 314,880 B
    qkv_attn_kernel<<<dim3(2 * HDIM * 16), dim3(256), smem, stream>>>(
        x, wq, bq, wk, bk, wv, bv, attn3);
    proj_kernel<<<dim3(16000), dim3(256), 0, stream>>>(attn3, wp, bp, out);
}